// CAT_1460288881350
// MI455X (gfx1250) — compile-verified
//
#include <hip/hip_runtime.h>
#include <hip/hip_bf16.h>
#include <math.h>

// Problem constants (match reference)
#define NFEAT   512
#define HIDDEN  512
#define NCLUS   16

typedef float v2f __attribute__((ext_vector_type(2)));
typedef float v8f __attribute__((ext_vector_type(8)));

__device__ __forceinline__ void atomAddF(float* p, float v) {
  // lowers to global_atomic_add_f32 (no CAS loop)
  unsafeAtomicAdd(p, v);
}

// ---------------------------------------------------------------------------
// Kernel 1: XW = X @ W   (fp32, native V_WMMA_F32_16X16X4_F32)
// One wave per 16-row x 64-col output strip; K loop in steps of 4.
// ---------------------------------------------------------------------------
__global__ void __launch_bounds__(32)
gemm_xw_wmma(const float* __restrict__ X, const float* __restrict__ W,
             float* __restrict__ XW, int nrows) {
  const int lane = threadIdx.x & 31;
  const int m    = lane & 15;
  const int kh   = lane >> 4;              // 0 or 1 (K half)
  const int row0 = blockIdx.x * 16;
  const int col0 = blockIdx.y * 64;

  int arow_idx = row0 + m;
  if (arow_idx >= nrows) arow_idx = nrows - 1;   // N is a multiple of 16; safety
  const float* arow = X + (size_t)arow_idx * NFEAT + 2 * kh;

  v8f acc0 = {}, acc1 = {}, acc2 = {}, acc3 = {};

  for (int k = 0; k < NFEAT; k += 4) {
    v2f a = *(const v2f*)(arow + k);                       // global_load_b64
    const float* b0 = W + (size_t)(k + 2 * kh) * HIDDEN + col0 + m;
    v2f bv0 = { b0[ 0], b0[HIDDEN +  0] };
    v2f bv1 = { b0[16], b0[HIDDEN + 16] };
    v2f bv2 = { b0[32], b0[HIDDEN + 32] };
    v2f bv3 = { b0[48], b0[HIDDEN + 48] };
    acc0 = __builtin_amdgcn_wmma_f32_16x16x4_f32(false, a, false, bv0, (short)0, acc0, false, false);
    acc1 = __builtin_amdgcn_wmma_f32_16x16x4_f32(false, a, false, bv1, (short)0, acc1, false, false);
    acc2 = __builtin_amdgcn_wmma_f32_16x16x4_f32(false, a, false, bv2, (short)0, acc2, false, false);
    acc3 = __builtin_amdgcn_wmma_f32_16x16x4_f32(false, a, false, bv3, (short)0, acc3, false, false);
  }

  // C/D layout: VGPR r -> (M = r + 8*kh, N = m)
  #pragma unroll
  for (int r = 0; r < 8; ++r) {
    int row = row0 + r + 8 * kh;
    if (row < nrows) {
      float* out = XW + (size_t)row * HIDDEN + col0 + m;
      out[ 0] = acc0[r];
      out[16] = acc1[r];
      out[32] = acc2[r];
      out[48] = acc3[r];
    }
  }
}

// ---------------------------------------------------------------------------
// CSR build, step 1: histogram of src + degrees (= segment_sum(gvals, dst))
// ---------------------------------------------------------------------------
__global__ void __launch_bounds__(256)
count_src_deg(const int* __restrict__ src, const int* __restrict__ dst,
              const float* __restrict__ gvals, int* __restrict__ counts,
              float* __restrict__ degrees, int E) {
  int e = blockIdx.x * blockDim.x + threadIdx.x;
  int stride = gridDim.x * blockDim.x;
  for (; e < E; e += stride) {
    atomicAdd(&counts[src[e]], 1);            // global_atomic_add_u32
    atomAddF(&degrees[dst[e]], gvals[e]);     // global_atomic_add_f32
  }
}

// ---------------------------------------------------------------------------
// CSR build, step 2: single-block exclusive scan of counts -> rowptr, cursor
// ---------------------------------------------------------------------------
__global__ void __launch_bounds__(1024)
scan_counts(const int* __restrict__ counts, int* __restrict__ rowptr,
            int* __restrict__ cursor, int N) {
  __shared__ int chunkSums[1024];
  const int tid   = threadIdx.x;
  const int chunk = (N + 1023) / 1024;
  const int start = tid * chunk;
  const int end   = min(start + chunk, N);

  int s = 0;
  for (int i = start; i < end; ++i) s += counts[i];
  chunkSums[tid] = s;
  __syncthreads();

  // Hillis-Steele inclusive scan over 1024 chunk sums
  for (int off = 1; off < 1024; off <<= 1) {
    int v = (tid >= off) ? chunkSums[tid - off] : 0;
    __syncthreads();
    chunkSums[tid] += v;
    __syncthreads();
  }

  int base = (tid == 0) ? 0 : chunkSums[tid - 1];
  for (int i = start; i < end; ++i) {
    rowptr[i] = base;
    cursor[i] = base;
    base += counts[i];
  }
  if (tid == 1023) rowptr[N] = chunkSums[1023];
}

// ---------------------------------------------------------------------------
// CSR build, step 3: bucket fill (edge ids grouped by src)
// ---------------------------------------------------------------------------
__global__ void __launch_bounds__(256)
fill_eidx(const int* __restrict__ src, int* __restrict__ cursor,
          int* __restrict__ eidx, int E) {
  int e = blockIdx.x * blockDim.x + threadIdx.x;
  int stride = gridDim.x * blockDim.x;
  for (; e < E; e += stride) {
    int pos = atomicAdd(&cursor[src[e]], 1);
    eidx[pos] = e;
  }
}

// ---------------------------------------------------------------------------
// Kernel 2': prop[n] = sum_{e in row n} gnorm[e] * XW[dst[e]]
// One wave per node; 512-float row accumulated in 16 VGPRs/lane; zero atomics.
// ---------------------------------------------------------------------------
__global__ void __launch_bounds__(256)
spmm_gather(const int* __restrict__ rowptr, const int* __restrict__ eidx,
            const int* __restrict__ dst, const float* __restrict__ gnorm,
            const float* __restrict__ XW, float* __restrict__ prop, int N) {
  const int lane = threadIdx.x & 31;
  const int n    = (int)((blockIdx.x * blockDim.x + threadIdx.x) >> 5);
  if (n >= N) return;

  const int start = rowptr[n];
  const int end   = rowptr[n + 1];

  float4 a0 = {0,0,0,0}, a1 = {0,0,0,0}, a2 = {0,0,0,0}, a3 = {0,0,0,0};

  for (int i = start; i < end; ++i) {
    const int   e = eidx[i];
    const int   d = dst[e];
    const float g = gnorm[e];
    if (i + 1 < end) {   // prefetch next gathered row (global_prefetch_b8)
      int d2 = dst[eidx[i + 1]];
      __builtin_prefetch(XW + (size_t)d2 * HIDDEN + lane * 4, 0, 1);
    }
    const float4* in = (const float4*)(XW + (size_t)d * HIDDEN) + lane;
    float4 v0 = in[0], v1 = in[32], v2 = in[64], v3 = in[96];  // b128 x4
    a0.x += g * v0.x; a0.y += g * v0.y; a0.z += g * v0.z; a0.w += g * v0.w;
    a1.x += g * v1.x; a1.y += g * v1.y; a1.z += g * v1.z; a1.w += g * v1.w;
    a2.x += g * v2.x; a2.y += g * v2.y; a2.z += g * v2.z; a2.w += g * v2.w;
    a3.x += g * v3.x; a3.y += g * v3.y; a3.z += g * v3.z; a3.w += g * v3.w;
  }

  float4* out = (float4*)(prop + (size_t)n * HIDDEN) + lane;
  out[0]  = a0;
  out[32] = a1;
  out[64] = a2;
  out[96] = a3;
}

// ---------------------------------------------------------------------------
// Kernel 3: assignments = softmax(selu(prop + b) @ Wt + bt)   (thread / node)
// ---------------------------------------------------------------------------
__global__ void __launch_bounds__(256)
assign_softmax(const float* __restrict__ prop, const float* __restrict__ bvec,
               const float* __restrict__ Wt, const float* __restrict__ bt,
               float* __restrict__ assign, int N) {
  const int n = blockIdx.x * blockDim.x + threadIdx.x;
  if (n >= N) return;
  const float* row = prop + (size_t)n * HIDDEN;

  float acc[NCLUS];
  #pragma unroll
  for (int k = 0; k < NCLUS; ++k) acc[k] = bt[k];

  const float kScale = 1.0507009873554805f;
  const float kAlpha = 1.6732632423543772f;
  for (int h = 0; h < HIDDEN; ++h) {
    float x = row[h] + bvec[h];
    float g = (x > 0.0f) ? kScale * x : kScale * kAlpha * (__expf(x) - 1.0f);
    const float* wt = Wt + h * NCLUS;
    #pragma unroll
    for (int k = 0; k < NCLUS; ++k) acc[k] = fmaf(g, wt[k], acc[k]);
  }

  float mx = acc[0];
  #pragma unroll
  for (int k = 1; k < NCLUS; ++k) mx = fmaxf(mx, acc[k]);
  float sum = 0.0f;
  #pragma unroll
  for (int k = 0; k < NCLUS; ++k) { acc[k] = __expf(acc[k] - mx); sum += acc[k]; }
  const float inv = 1.0f / sum;
  float* outp = assign + (size_t)n * NCLUS;
  #pragma unroll
  for (int k = 0; k < NCLUS; ++k) outp[k] = acc[k] * inv;
}

// ---------------------------------------------------------------------------
// Kernel 4': As[n] = sum_{e in row n} gvals[e] * assign[dst[e]]
// One wave per node; lanes 0-15 / 16-31 process alternating edges, folded by
// a cross-half shuffle.  Zero atomics; As written unconditionally.
// ---------------------------------------------------------------------------
__global__ void __launch_bounds__(256)
as_gather(const int* __restrict__ rowptr, const int* __restrict__ eidx,
          const int* __restrict__ dst, const float* __restrict__ gvals,
          const float* __restrict__ assign, float* __restrict__ As, int N) {
  const int lane = threadIdx.x & 31;
  const int k    = lane & 15;
  const int half = lane >> 4;
  const int n    = (int)((blockIdx.x * blockDim.x + threadIdx.x) >> 5);
  if (n >= N) return;

  const int start = rowptr[n];
  const int end   = rowptr[n + 1];

  float acc = 0.0f;
  for (int i = start + half; i < end; i += 2) {
    const int e = eidx[i];
    acc += gvals[e] * assign[(size_t)dst[e] * NCLUS + k];
  }
  acc += __shfl_down(acc, 16, 32);    // fold upper half into lower
  if (lane < NCLUS) As[(size_t)n * NCLUS + k] = acc;
}

// ---------------------------------------------------------------------------
// Kernel 5: per-node reduction into 33 global accumulators
//   acc[0]      = trace(As^T @ assign)
//   acc[1..16]  = normalizer_left[k]
//   acc[17..32] = cluster_sizes[k]
// ---------------------------------------------------------------------------
__global__ void __launch_bounds__(256)
node_reduce(const float* __restrict__ assign, const float* __restrict__ As,
            const float* __restrict__ degrees, float* __restrict__ acc, int N) {
  __shared__ float s_tr;
  __shared__ float s_nl[NCLUS];
  __shared__ float s_cs[NCLUS];
  const int tid = threadIdx.x;
  if (tid == 0) s_tr = 0.0f;
  if (tid < NCLUS) { s_nl[tid] = 0.0f; s_cs[tid] = 0.0f; }
  __syncthreads();

  const int n = blockIdx.x * blockDim.x + tid;
  if (n < N) {
    const float deg = degrees[n];
    float tr = 0.0f;
    #pragma unroll
    for (int k = 0; k < NCLUS; ++k) {
      const float a = assign[(size_t)n * NCLUS + k];
      tr += a * As[(size_t)n * NCLUS + k];
      atomicAdd(&s_nl[k], a * deg);   // ds_add_f32
      atomicAdd(&s_cs[k], a);         // ds_add_f32
    }
    atomicAdd(&s_tr, tr);
  }
  __syncthreads();

  if (tid == 0) atomAddF(&acc[0], s_tr);
  if (tid < NCLUS) {
    atomAddF(&acc[1 + tid], s_nl[tid]);
    atomAddF(&acc[1 + NCLUS + tid], s_cs[tid]);
  }
}

// ---------------------------------------------------------------------------
// Kernel 6: finalize scalar loss
// ---------------------------------------------------------------------------
__global__ void finalize_loss(const float* __restrict__ acc, float* __restrict__ out,
                              float Ef, float Nf) {
  if (blockIdx.x == 0 && threadIdx.x == 0) {
    const float tr = acc[0];
    float nrm = 0.0f, cs2 = 0.0f;
    #pragma unroll
    for (int k = 0; k < NCLUS; ++k) {
      const float nl = acc[1 + k];
      const float cs = acc[1 + NCLUS + k];
      nrm += nl * nl;
      cs2 += cs * cs;
    }
    nrm = nrm / (2.0f * Ef);                       // trace(normalizer)
    const float spectral = -(tr - nrm) / (2.0f * Ef);
    const float cluster  = (sqrtf(cs2) / Nf * 4.0f - 1.0f) * 1.0f; // sqrt(16)=4
    out[0] = spectral + cluster;
  }
}

// ---------------------------------------------------------------------------
// Host launcher
// ---------------------------------------------------------------------------
extern "C" void kernel_launch(void* const* d_in, const int* in_sizes, int n_in,
                              void* d_out, int out_size, void* d_ws, size_t ws_size,
                              hipStream_t stream) {
  const int*   src   = (const int*)d_in[0];
  const int*   dst   = (const int*)d_in[1];
  const float* gvals = (const float*)d_in[2];
  const float* gnorm = (const float*)d_in[3];
  const float* X     = (const float*)d_in[4];
  const float* W     = (const float*)d_in[5];
  const float* bvec  = (const float*)d_in[6];
  const float* Wt    = (const float*)d_in[7];
  const float* bt    = (const float*)d_in[8];

  const int E = in_sizes[0];
  const int N = in_sizes[4] / NFEAT;

  // Workspace layout (256B aligned)
  char*  ws  = (char*)d_ws;
  size_t off = 0;
  auto align256 = [](size_t x) { return (x + 255) & ~(size_t)255; };

  float* xw     = (float*)(ws + off); off = align256(off + (size_t)N * HIDDEN * 4);
  float* prop   = (float*)(ws + off); off = align256(off + (size_t)N * HIDDEN * 4);
  float* assign = (float*)(ws + off); off = align256(off + (size_t)N * NCLUS * 4);
  float* As     = (float*)(ws + off); off = align256(off + (size_t)N * NCLUS * 4);
  // --- zero-initialized region: counts, degrees, acc ---
  size_t zOff   = off;
  int*   counts = (int*)  (ws + off); off = align256(off + (size_t)N * 4);
  float* deg    = (float*)(ws + off); off = align256(off + (size_t)N * 4);
  float* acc    = (float*)(ws + off); off = align256(off + 33 * 4);
  size_t zEnd   = off;
  // --- CSR index ---
  int*   rowptr = (int*)(ws + off); off = align256(off + (size_t)(N + 1) * 4);
  int*   cursor = (int*)(ws + off); off = align256(off + (size_t)N * 4);
  int*   eidx   = (int*)(ws + off); off = align256(off + (size_t)E * 4);

  // Zero only the small accumulation region (prop/As are fully overwritten).
  hipMemsetAsync(ws + zOff, 0, zEnd - zOff, stream);

  const int nodeWaveBlocks = (N * 32 + 255) / 256;   // one wave per node

  // CSR build
  count_src_deg<<<2048, 256, 0, stream>>>(src, dst, gvals, counts, deg, E);
  scan_counts<<<1, 1024, 0, stream>>>(counts, rowptr, cursor, N);
  fill_eidx<<<2048, 256, 0, stream>>>(src, cursor, eidx, E);

  // 1) XW = X @ W via fp32 WMMA
  {
    dim3 grid((N + 15) / 16, HIDDEN / 64);
    gemm_xw_wmma<<<grid, 32, 0, stream>>>(X, W, xw, N);
  }
  // 2) sparse propagate: pure gather, no atomics
  spmm_gather<<<nodeWaveBlocks, 256, 0, stream>>>(rowptr, eidx, dst, gnorm, xw, prop, N);
  // 3) selu + transform + softmax
  assign_softmax<<<(N + 255) / 256, 256, 0, stream>>>(prop, bvec, Wt, bt, assign, N);
  // 4) pooled gather: A @ S rows, no atomics
  as_gather<<<nodeWaveBlocks, 256, 0, stream>>>(rowptr, eidx, dst, gvals, assign, As, N);
  // 5) node-level reduction
  node_reduce<<<(N + 255) / 256, 256, 0, stream>>>(assign, As, deg, acc, N);
  // 6) scalar loss
  finalize_loss<<<1, 32, 0, stream>>>(acc, (float*)d_out, (float)E, (float)N);
}